// GatedGCN_33904471835040
// MI455X (gfx1250) — compile-verified
//
#include <hip/hip_runtime.h>
#include <hip/hip_bf16.h>
#include <math.h>

typedef __attribute__((ext_vector_type(2))) float v2f;
typedef __attribute__((ext_vector_type(8))) float v8f;

#define H 128
#define O3 384  // 3*H

__device__ __forceinline__ v8f wmma_f32(v2f a, v2f b, v8f c) {
    // D = A(16x4 f32) * B(4x16 f32) + C(16x16 f32)
    return __builtin_amdgcn_wmma_f32_16x16x4_f32(
        false, a, false, b, (short)0, c, false, false);
}

__device__ __forceinline__ float sigmoidf_(float x) {
    return 1.0f / (1.0f + expf(-x));
}

// ---------------- utility kernels ----------------

__global__ void k_copy(const float* __restrict__ src, float* __restrict__ dst, long n) {
    long i = (long)blockIdx.x * blockDim.x + threadIdx.x;
    long stride = (long)gridDim.x * blockDim.x;
    for (; i < n; i += stride) dst[i] = src[i];
}

__global__ void k_zero(float* __restrict__ p, long n) {
    long i = (long)blockIdx.x * blockDim.x + threadIdx.x;
    long stride = (long)gridDim.x * blockDim.x;
    for (; i < n; i += stride) p[i] = 0.0f;
}

// ---------------- m = h @ W_l  (N x 128 @ 128 x 128, B not transposed) ----------------
// block = 256 threads (8 waves), one block per 32-node tile (two 16-row M-tiles).
// Wave w computes output columns [w*16, w*16+16) for BOTH M-tiles, giving two
// independent WMMA accumulation chains that share one B fragment per k-step.
__global__ __launch_bounds__(256) void k_gemm_msg(
    const float* __restrict__ h, const float* __restrict__ W, float* __restrict__ m, int N)
{
    __shared__ float Ah[32 * 132];
    const int node0 = blockIdx.x * 32;

    for (int i = threadIdx.x; i < 32 * H; i += 256) {
        int r = i >> 7, c = i & 127;
        int row = node0 + r; if (row >= N) row = N - 1;
        Ah[r * 132 + c] = h[(long)row * H + c];
    }
    __syncthreads();

    const int wv   = threadIdx.x >> 5;
    const int lane = threadIdx.x & 31;
    const int mrow = lane & 15;
    const int khalf = (lane >> 4) << 1;   // 0 or 2
    const int ncol = wv * 16 + (lane & 15);

    v8f acc0 = {};
    v8f acc1 = {};
    #pragma unroll 8
    for (int ks = 0; ks < 32; ++ks) {
        int kk = ks * 4 + khalf;
        v2f a0 = *(const v2f*)&Ah[mrow * 132 + kk];
        v2f a1 = *(const v2f*)&Ah[(16 + mrow) * 132 + kk];
        v2f b;
        b.x = W[kk * H + ncol];
        b.y = W[(kk + 1) * H + ncol];
        acc0 = wmma_f32(a0, b, acc0);
        acc1 = wmma_f32(a1, b, acc1);
    }

    const int rbase = (lane >> 4) << 3;   // 0 or 8
    if (node0 + 32 <= N) {
        #pragma unroll
        for (int r = 0; r < 8; ++r) {
            m[(long)(node0 + rbase + r) * H + ncol]      = acc0[r];
            m[(long)(node0 + 16 + rbase + r) * H + ncol] = acc1[r];
        }
    } else {
        #pragma unroll
        for (int r = 0; r < 8; ++r) {
            int row0 = node0 + rbase + r;
            if (row0 < N) m[(long)row0 * H + ncol] = acc0[r];
            int row1 = node0 + 16 + rbase + r;
            if (row1 < N) m[(long)row1 * H + ncol] = acc1[r];
        }
    }
}

// ---------------- scatter: agg[dst] += m[src] * ew, one wave per edge ----------------
__global__ __launch_bounds__(256) void k_scatter(
    const float* __restrict__ m, const long long* __restrict__ ei,
    const float* __restrict__ ew, float* __restrict__ agg, long E)
{
    long gid  = (long)blockIdx.x * blockDim.x + threadIdx.x;
    long wave = gid >> 5;
    int  lane = (int)(gid & 31);
    long nwaves = ((long)gridDim.x * blockDim.x) >> 5;

    for (long e = wave; e < E; e += nwaves) {
        long s = ei[e];
        long d = ei[E + e];
        float w = ew[e];
        const float4* srcv = (const float4*)(m + s * H);
        float4 v = srcv[lane];
        float* dstp = agg + d * H + lane * 4;
        atomicAdd(dstp + 0, v.x * w);
        atomicAdd(dstp + 1, v.y * w);
        atomicAdd(dstp + 2, v.z * w);
        atomicAdd(dstp + 3, v.w * w);
    }
}

// ---------------- fused GRU layer ----------------
// gi = agg @ W_ih^T ; gh = h @ W_hh^T ; gates ; h <- (1-z)*n + z*h   (in place)
// block = 256 threads (8 waves) per 16-node tile. Wave w computes gi/gh output
// tiles for columns [w*48, w*48+48) (3 tiles each of gi and gh; the two WMMA
// accumulation chains per tile-pair are interleaved for XDL latency hiding).
__global__ __launch_bounds__(256) void k_gru_fused(
    float* __restrict__ hio, const float* __restrict__ agg,
    const float* __restrict__ w_ih, const float* __restrict__ w_hh,
    const float* __restrict__ b_ih, const float* __restrict__ b_hh, int N)
{
    __shared__ float Aagg[16 * 132];
    __shared__ float Ah[16 * 132];
    __shared__ float Gi[16 * O3];
    __shared__ float Gh[16 * O3];

    const int node0 = blockIdx.x * 16;

    for (int i = threadIdx.x; i < 16 * H; i += 256) {
        int r = i >> 7, c = i & 127;
        int row = node0 + r; if (row >= N) row = N - 1;
        Aagg[r * 132 + c] = agg[(long)row * H + c];
        Ah[r * 132 + c]   = hio[(long)row * H + c];
    }
    __syncthreads();

    const int wv   = threadIdx.x >> 5;
    const int lane = threadIdx.x & 31;
    const int mrow = lane & 15;
    const int khalf = (lane >> 4) << 1;
    const int rbase = (lane >> 4) << 3;

    #pragma unroll
    for (int t = 0; t < 3; ++t) {
        const int tn = wv * 3 + t;                 // output 16-col tile in [0,24)
        const int ncol = tn * 16 + (lane & 15);    // column in [0,384)
        v8f cg = {};
        v8f ch = {};
        #pragma unroll 8
        for (int ks = 0; ks < 32; ++ks) {
            int kk = ks * 4 + khalf;
            v2f aA = *(const v2f*)&Aagg[mrow * 132 + kk];
            v2f aH = *(const v2f*)&Ah[mrow * 132 + kk];
            // W is (384 x 128) row-major; B(k,n) = W[n*128 + k] (transposed use)
            v2f bI = *(const v2f*)&w_ih[(long)ncol * H + kk];
            v2f bH = *(const v2f*)&w_hh[(long)ncol * H + kk];
            cg = wmma_f32(aA, bI, cg);
            ch = wmma_f32(aH, bH, ch);
        }
        #pragma unroll
        for (int r = 0; r < 8; ++r) {
            Gi[(rbase + r) * O3 + ncol] = cg[r];
            Gh[(rbase + r) * O3 + ncol] = ch[r];
        }
    }
    __syncthreads();

    if (node0 + 16 <= N) {
        for (int i = threadIdx.x; i < 16 * H; i += 256) {
            int r = i >> 7, f = i & 127;
            float ir  = Gi[r * O3 + f]          + b_ih[f];
            float iz  = Gi[r * O3 + H + f]      + b_ih[H + f];
            float in_ = Gi[r * O3 + 2 * H + f]  + b_ih[2 * H + f];
            float hr  = Gh[r * O3 + f]          + b_hh[f];
            float hz  = Gh[r * O3 + H + f]      + b_hh[H + f];
            float hn  = Gh[r * O3 + 2 * H + f]  + b_hh[2 * H + f];
            float rr = sigmoidf_(ir + hr);
            float zz = sigmoidf_(iz + hz);
            float nn = tanhf(in_ + rr * hn);
            float hv = Ah[r * 132 + f];
            hio[(long)(node0 + r) * H + f] = (1.0f - zz) * nn + zz * hv;
        }
    } else {
        for (int i = threadIdx.x; i < 16 * H; i += 256) {
            int r = i >> 7, f = i & 127;
            int row = node0 + r;
            if (row >= N) continue;
            float ir  = Gi[r * O3 + f]          + b_ih[f];
            float iz  = Gi[r * O3 + H + f]      + b_ih[H + f];
            float in_ = Gi[r * O3 + 2 * H + f]  + b_ih[2 * H + f];
            float hr  = Gh[r * O3 + f]          + b_hh[f];
            float hz  = Gh[r * O3 + H + f]      + b_hh[H + f];
            float hn  = Gh[r * O3 + 2 * H + f]  + b_hh[2 * H + f];
            float rr = sigmoidf_(ir + hr);
            float zz = sigmoidf_(iz + hz);
            float nn = tanhf(in_ + rr * hn);
            float hv = Ah[r * 132 + f];
            hio[(long)row * H + f] = (1.0f - zz) * nn + zz * hv;
        }
    }
}

// ---------------- mean pooling over batch ----------------
__global__ void k_pool_acc(const float* __restrict__ h, const long long* __restrict__ batch,
                           float* __restrict__ sums, float* __restrict__ cnts, long NH)
{
    long i = (long)blockIdx.x * blockDim.x + threadIdx.x;
    long stride = (long)gridDim.x * blockDim.x;
    for (; i < NH; i += stride) {
        long n = i >> 7;
        int f = (int)(i & 127);
        long g = batch[n];
        atomicAdd(&sums[g * H + f], h[i]);
        if (f == 0) atomicAdd(&cnts[g], 1.0f);
    }
}

__global__ void k_pool_div(const float* __restrict__ sums, const float* __restrict__ cnts,
                           float* __restrict__ pooled, int GH)
{
    int i = blockIdx.x * blockDim.x + threadIdx.x;
    if (i < GH) pooled[i] = sums[i] / fmaxf(cnts[i >> 7], 1.0f);
}

// ---------------- tiny dense layers: out = act(in @ W^T + b) ----------------
__global__ void k_dense(const float* __restrict__ in, const float* __restrict__ W,
                        const float* __restrict__ b, float* __restrict__ out,
                        int G, int O, int relu)
{
    int i = blockIdx.x * blockDim.x + threadIdx.x;
    if (i >= G * O) return;
    int g = i / O, o = i % O;
    float acc = b[o];
    const float* ip = in + (long)g * H;
    const float* wp = W + (long)o * H;
    #pragma unroll 8
    for (int k = 0; k < H; ++k) acc += ip[k] * wp[k];
    out[i] = relu ? fmaxf(acc, 0.0f) : acc;
}

// ---------------- launcher ----------------
extern "C" void kernel_launch(void* const* d_in, const int* in_sizes, int n_in,
                              void* d_out, int out_size, void* d_ws, size_t ws_size,
                              hipStream_t stream) {
    const float*     x        = (const float*)d_in[0];
    const long long* ei       = (const long long*)d_in[1];
    const long long* batch    = (const long long*)d_in[2];
    const float*     ew       = (const float*)d_in[3];
    const float*     W_msg    = (const float*)d_in[4];
    const float*     gru_w_ih = (const float*)d_in[5];
    const float*     gru_w_hh = (const float*)d_in[6];
    const float*     gru_b_ih = (const float*)d_in[7];
    const float*     gru_b_hh = (const float*)d_in[8];
    const float*     W_in     = (const float*)d_in[9];
    const float*     b_in     = (const float*)d_in[10];
    const float*     W_d1     = (const float*)d_in[11];
    const float*     b_d1     = (const float*)d_in[12];
    const float*     W_d2     = (const float*)d_in[13];
    const float*     b_d2     = (const float*)d_in[14];
    const float*     W_out    = (const float*)d_in[15];
    const float*     b_out    = (const float*)d_in[16];

    const int  N = in_sizes[0] / H;
    const long E = (long)in_sizes[3];
    const int  L = in_sizes[4] / (H * H);
    const int  C = in_sizes[15] / H;
    const int  G = 64;
    const long NH = (long)N * H;

    float* ws = (float*)d_ws;
    float* h      = ws;            ws += NH;
    float* m      = ws;            ws += NH;
    float* agg    = ws;            ws += NH;
    float* sums   = ws;            ws += (long)G * H;
    float* cnts   = ws;            ws += G;
    float* pooled = ws;            ws += (long)G * H;
    float* z1     = ws;            ws += (long)G * H;
    float* z2     = ws;            ws += (long)G * H;
    float* z3     = ws;            ws += (long)G * H;

    const int Mtiles16 = (N + 15) / 16;
    const int Mtiles32 = (N + 31) / 32;

    k_copy<<<2048, 256, 0, stream>>>(x, h, NH);

    for (int l = 0; l < L; ++l) {
        k_gemm_msg<<<Mtiles32, 256, 0, stream>>>(h, W_msg + (long)l * H * H, m, N);
        k_zero<<<2048, 256, 0, stream>>>(agg, NH);
        k_scatter<<<8192, 256, 0, stream>>>(m, ei, ew, agg, E);
        k_gru_fused<<<Mtiles16, 256, 0, stream>>>(h, agg, gru_w_ih, gru_w_hh,
                                                  gru_b_ih, gru_b_hh, N);
    }

    k_zero<<<64, 256, 0, stream>>>(sums, (long)G * H);
    k_zero<<<1, 64, 0, stream>>>(cnts, G);
    k_pool_acc<<<2048, 256, 0, stream>>>(h, batch, sums, cnts, NH);
    k_pool_div<<<(G * H + 255) / 256, 256, 0, stream>>>(sums, cnts, pooled, G * H);

    k_dense<<<(G * H + 255) / 256, 256, 0, stream>>>(pooled, W_in, b_in, z1, G, H, 1);
    k_dense<<<(G * H + 255) / 256, 256, 0, stream>>>(z1, W_d1, b_d1, z2, G, H, 1);
    k_dense<<<(G * H + 255) / 256, 256, 0, stream>>>(z2, W_d2, b_d2, z3, G, H, 1);
    k_dense<<<(G * C + 255) / 256, 256, 0, stream>>>(z3, W_out, b_out, (float*)d_out, G, C, 0);
}